// Kalman_Smooth_Gradient_29119878267470
// MI455X (gfx1250) — compile-verified
//
#include <hip/hip_runtime.h>
#include <math.h>

// ---------------------------------------------------------------------------
// Batched EKF forward + RTS smoother + innovation loss, one sequence per lane.
// B = 4096 sequences, T = 128 steps, 6-state, 3-obs.
//
// Workspace layout (floats), total ~63 MB:
//   [0)                : zt   = transposed measurements, [T*3][B]
//   [384*B)            : traj = per-step {s_f(6), upper(P_f)(21)}, comp-major
//   [384*B + 27*T*B)   : wave partial losses (B/32 = 128 floats)
// ---------------------------------------------------------------------------

#define B_N 4096
#define T_N 128
#define DT_ (1.0f / 120.0f)
#define TWO_PI_F 6.28318530717958647692f
#define INV_TWO_PI_F 0.15915494309189533577f
#define THR_15PI 4.71238898038468985769f  // 1.5*pi

typedef __attribute__((ext_vector_type(2))) float v2f;
typedef __attribute__((ext_vector_type(8))) float v8f;

#if defined(__has_builtin)
#if __has_builtin(__builtin_amdgcn_wmma_f32_16x16x4_f32)
#define HAVE_WMMA_F32X4 1
#endif
#endif
#ifndef HAVE_WMMA_F32X4
#define HAVE_WMMA_F32X4 0
#endif

// ---------------------------------------------------------------------------
// z transpose: [B][T*3] -> [T*3][B] so the EKF kernel's per-step measurement
// loads are lane-contiguous (coalesced). Classic padded-LDS tile transpose.
// ---------------------------------------------------------------------------
__global__ __launch_bounds__(256) void transpose_z_kernel(
    const float* __restrict__ z, float* __restrict__ zt) {
  __shared__ float tile[32][33];
  const int c0 = blockIdx.x * 32;  // column tile (c = t*3+k, 0..383)
  const int b0 = blockIdx.y * 32;  // row tile (b)
  const int tx = threadIdx.x;      // 0..31
  const int ty = threadIdx.y;      // 0..7
#pragma unroll
  for (int r = 0; r < 4; ++r) {
    const int br = ty + r * 8;
    tile[br][tx] = z[(size_t)(b0 + br) * (T_N * 3) + (c0 + tx)];
  }
  __syncthreads();
#pragma unroll
  for (int r = 0; r < 4; ++r) {
    const int cr = ty + r * 8;
    zt[(size_t)(c0 + cr) * B_N + (b0 + tx)] = tile[tx][cr];
  }
}

// ---------------------------------------------------------------------------
// Small-matrix helpers. F is I6 with F[0][2]=F[1][3]=F[4][5]=DT,
// F[2][2]=gx, F[3][3]=gy, F[5][5]=g  -> apply it sparsely.
// ---------------------------------------------------------------------------
__device__ __forceinline__ void f_left(const float P[6][6], float A[6][6],
                                       float gx, float gy, float g) {
  // A = F * P (row operations)
#pragma unroll
  for (int j = 0; j < 6; ++j) {
    A[0][j] = fmaf(DT_, P[2][j], P[0][j]);
    A[1][j] = fmaf(DT_, P[3][j], P[1][j]);
    A[2][j] = gx * P[2][j];
    A[3][j] = gy * P[3][j];
    A[4][j] = fmaf(DT_, P[5][j], P[4][j]);
    A[5][j] = g * P[5][j];
  }
}

__device__ __forceinline__ void f_rightT(const float A[6][6], float Pp[6][6],
                                         float gx, float gy, float g) {
  // Pp = A * F^T (column operations)
#pragma unroll
  for (int i = 0; i < 6; ++i) {
    Pp[i][0] = fmaf(DT_, A[i][2], A[i][0]);
    Pp[i][1] = fmaf(DT_, A[i][3], A[i][1]);
    Pp[i][2] = gx * A[i][2];
    Pp[i][3] = gy * A[i][3];
    Pp[i][4] = fmaf(DT_, A[i][5], A[i][4]);
    Pp[i][5] = g * A[i][5];
  }
}

__device__ __forceinline__ void predict_state(const float s[6], float sp[6],
                                              float& gx, float& gy,
                                              float friction, float damping,
                                              float g) {
  const float dx = s[2], dy = s[3];
  const float tx = tanhf(100.0f * dx);
  const float ty = tanhf(100.0f * dy);
  sp[0] = fmaf(DT_, s[2], s[0]);
  sp[1] = fmaf(DT_, s[3], s[1]);
  sp[2] = dx - DT_ * (damping * dx + friction * tx);
  sp[3] = dy - DT_ * (damping * dy + friction * ty);
  sp[4] = fmaf(DT_, s[5], s[4]);
  sp[5] = s[5] * g;
  gx = 1.0f - DT_ * (damping + friction * 100.0f * (1.0f - tx * tx));
  gy = 1.0f - DT_ * (damping + friction * 100.0f * (1.0f - ty * ty));
}

// Solve Pp * X = Afp for SPD Pp via Cholesky, return G = X^T.
__device__ __forceinline__ void spd_solve_T(const float Pp[6][6],
                                            const float Afp[6][6],
                                            float G[6][6]) {
  float L[6][6];
  float invd[6];
#pragma unroll
  for (int j = 0; j < 6; ++j) {
    float d = Pp[j][j];
#pragma unroll
    for (int k = 0; k < j; ++k) d = fmaf(-L[j][k], L[j][k], d);
    const float lj = sqrtf(fmaxf(d, 1e-30f));
    L[j][j] = lj;
    invd[j] = 1.0f / lj;
#pragma unroll
    for (int i = j + 1; i < 6; ++i) {
      float a = Pp[i][j];
#pragma unroll
      for (int k = 0; k < j; ++k) a = fmaf(-L[i][k], L[j][k], a);
      L[i][j] = a * invd[j];
    }
  }
  float X[6][6];
#pragma unroll
  for (int c = 0; c < 6; ++c) {
#pragma unroll
    for (int i = 0; i < 6; ++i) {
      float a = Afp[i][c];
#pragma unroll
      for (int k = 0; k < i; ++k) a = fmaf(-L[i][k], X[k][c], a);
      X[i][c] = a * invd[i];
    }
#pragma unroll
    for (int i = 5; i >= 0; --i) {
      float a = X[i][c];
#pragma unroll
      for (int k = i + 1; k < 6; ++k) a = fmaf(-L[k][i], X[k][c], a);
      X[i][c] = a * invd[i];
    }
  }
#pragma unroll
  for (int i = 0; i < 6; ++i)
#pragma unroll
    for (int j = 0; j < 6; ++j) G[i][j] = X[j][i];
}

// det(Sv) + innov^T Sv innov with H-selected symmetric Pt entries.
__device__ __forceinline__ float loss_term(float z0, float z1, float z2,
                                           const float st[6],
                                           const float Pt[6][6], float R0,
                                           float R1, float R2) {
  const float S00 = Pt[0][0] + R0, S01 = Pt[0][1], S02 = Pt[0][4];
  const float S11 = Pt[1][1] + R1, S12 = Pt[1][4];
  const float S22 = Pt[4][4] + R2;
  const float ix = z0 - st[0];
  const float iy = z1 - st[1];
  float a = z2 - st[4];
  a = (a > THR_15PI) ? a - TWO_PI_F : a;
  a = (a < -THR_15PI) ? a + TWO_PI_F : a;
  const float det = S00 * (S11 * S22 - S12 * S12) -
                    S01 * (S01 * S22 - S12 * S02) +
                    S02 * (S01 * S12 - S11 * S02);
  const float q = ix * (S00 * ix + S01 * iy + S02 * a) +
                  iy * (S01 * ix + S11 * iy + S12 * a) +
                  a * (S02 * ix + S12 * iy + S22 * a);
  return det + q;
}

// ---------------------------------------------------------------------------
// Main kernel: one sequence per lane. Forward EKF (stores s_f + upper(P_f)),
// then backward RTS smoother with recomputed predictions, accumulating the
// loss on the fly. Per-wave loss is reduced through the WMMA pipe.
// ---------------------------------------------------------------------------
__global__ __launch_bounds__(256) void ekf_rts_kernel(
    const float* __restrict__ params, const float* __restrict__ covp,
    const float* __restrict__ init, const float* __restrict__ zt,
    float* __restrict__ traj, float* __restrict__ partials) {
  const int b = blockIdx.x * blockDim.x + threadIdx.x;  // 0..B_N-1 exactly

  // Scalar (wave-uniform) parameters — compiler lifts these to SGPRs.
  const float friction = (tanhf(params[0]) + 1.0f) * 0.01f;
  const float damping = (tanhf(params[1]) + 1.0f) * 0.01f;
  const float g = 1.0f - DT_ * damping;
  float cp[7];
#pragma unroll
  for (int i = 0; i < 7; ++i) cp[i] = 1.0f / (1.0f + expf(-covp[i]));
  const float R0 = cp[0], R1 = cp[1], R2 = cp[2];
  const float qd[6] = {cp[3], cp[3], cp[4], cp[4], cp[5], cp[6]};

  // ---- state init ----
  float s[6];
#pragma unroll
  for (int j = 0; j < 6; ++j) s[j] = init[(size_t)b * 6 + j];
  float P[6][6];
#pragma unroll
  for (int i = 0; i < 6; ++i)
#pragma unroll
    for (int j = 0; j < 6; ++j) P[i][j] = (i == j) ? 0.01f : 0.0f;

  // ---- forward EKF ----
  for (int t = 0; t < T_N; ++t) {
    if (t + 8 < T_N)  // prefetch upcoming measurement rows (global_prefetch_b8)
      __builtin_prefetch(&zt[((size_t)(t + 8) * 3) * B_N + b], 0, 3);

    float sp[6], gx, gy;
    predict_state(s, sp, gx, gy, friction, damping, g);

    float A[6][6], Pp[6][6];
    f_left(P, A, gx, gy, g);
    f_rightT(A, Pp, gx, gy, g);
#pragma unroll
    for (int i = 0; i < 6; ++i) Pp[i][i] += qd[i];

    const float z0 = zt[((size_t)t * 3 + 0) * B_N + b];
    const float z1 = zt[((size_t)t * 3 + 1) * B_N + b];
    const float z2 = zt[((size_t)t * 3 + 2) * B_N + b];

    // S = H Pp H^T + R with H selecting rows {0,1,4}
    const float S00 = Pp[0][0] + R0, S01 = Pp[0][1], S02 = Pp[0][4];
    const float S11 = Pp[1][1] + R1, S12 = Pp[1][4];
    const float S22 = Pp[4][4] + R2;

    const float i0 = z0 - sp[0];
    const float i1 = z1 - sp[1];
    float a2 = z2 - sp[4];
    a2 = a2 - TWO_PI_F * rintf(a2 * INV_TWO_PI_F);  // _wrap (half-to-even)

    // 3x3 symmetric inverse (adjugate / det)
    const float c00 = S11 * S22 - S12 * S12;
    const float c01 = -(S01 * S22 - S12 * S02);
    const float c02 = S01 * S12 - S11 * S02;
    const float det = S00 * c00 + S01 * c01 + S02 * c02;
    const float id = 1.0f / det;
    const float iS00 = c00 * id, iS01 = c01 * id, iS02 = c02 * id;
    const float iS11 = (S00 * S22 - S02 * S02) * id;
    const float iS12 = -(S00 * S12 - S01 * S02) * id;
    const float iS22 = (S00 * S11 - S01 * S01) * id;

    // K = (S^-1 (H Pp))^T : K[j][k] = iS row k · (Pp[0][j],Pp[1][j],Pp[4][j])
    float K[6][3];
#pragma unroll
    for (int j = 0; j < 6; ++j) {
      const float m0 = Pp[0][j], m1 = Pp[1][j], m2 = Pp[4][j];
      K[j][0] = iS00 * m0 + iS01 * m1 + iS02 * m2;
      K[j][1] = iS01 * m0 + iS11 * m1 + iS12 * m2;
      K[j][2] = iS02 * m0 + iS12 * m1 + iS22 * m2;
    }
#pragma unroll
    for (int j = 0; j < 6; ++j)
      s[j] = sp[j] + K[j][0] * i0 + K[j][1] * i1 + K[j][2] * a2;
    // P = (I - K H) Pp = Pp - K * (H Pp)
#pragma unroll
    for (int i = 0; i < 6; ++i)
#pragma unroll
      for (int j = 0; j < 6; ++j)
        P[i][j] = Pp[i][j] -
                  (K[i][0] * Pp[0][j] + K[i][1] * Pp[1][j] + K[i][2] * Pp[4][j]);

    // store s_f + upper(P_f), component-major (all stores lane-coalesced)
    const size_t base = (size_t)t * 27 * B_N + (size_t)b;
#pragma unroll
    for (int j = 0; j < 6; ++j) traj[base + (size_t)j * B_N] = s[j];
    int c = 6;
#pragma unroll
    for (int i = 0; i < 6; ++i)
#pragma unroll
      for (int j = i; j < 6; ++j) {
        traj[base + (size_t)c * B_N] = P[i][j];
        ++c;
      }
  }

  // ---- backward RTS + loss ----
  float loss = 0.0f;
  {  // t = T-1 term uses the filtered (== smoothed) last state
    const float z0 = zt[((size_t)(T_N - 1) * 3 + 0) * B_N + b];
    const float z1 = zt[((size_t)(T_N - 1) * 3 + 1) * B_N + b];
    const float z2 = zt[((size_t)(T_N - 1) * 3 + 2) * B_N + b];
    loss += loss_term(z0, z1, z2, s, P, R0, R1, R2);
  }
  float sn[6];
  float Pn[6][6];
#pragma unroll
  for (int j = 0; j < 6; ++j) sn[j] = s[j];
#pragma unroll
  for (int i = 0; i < 6; ++i)
#pragma unroll
    for (int j = 0; j < 6; ++j) Pn[i][j] = P[i][j];

  for (int n = T_N - 2; n >= 0; --n) {
    const size_t base = (size_t)n * 27 * B_N + (size_t)b;
    if (n >= 4)  // prefetch trajectory rows we'll need soon
      __builtin_prefetch(&traj[base - (size_t)4 * 27 * B_N], 0, 3);

    float sf[6], Pf[6][6];
#pragma unroll
    for (int j = 0; j < 6; ++j) sf[j] = traj[base + (size_t)j * B_N];
    {
      int c = 6;
#pragma unroll
      for (int i = 0; i < 6; ++i)
#pragma unroll
        for (int j = i; j < 6; ++j) {
          const float v = traj[base + (size_t)c * B_N];
          Pf[i][j] = v;
          Pf[j][i] = v;
          ++c;
        }
    }

    // Recompute prediction at n+1 from s_f[n], P_f[n] (F[n+1] = dF(s_f[n]))
    float sp[6], gx, gy;
    predict_state(sf, sp, gx, gy, friction, damping, g);
    float A[6][6], Pp[6][6];
    f_left(Pf, A, gx, gy, g);  // A = F @ P_f  (== RHS of the gain solve)
    f_rightT(A, Pp, gx, gy, g);
#pragma unroll
    for (int i = 0; i < 6; ++i) Pp[i][i] += qd[i];

    float G[6][6];
    spd_solve_T(Pp, A, G);  // G = (Pp^-1 (F P_f))^T

    float ds_[6];
#pragma unroll
    for (int j = 0; j < 6; ++j) ds_[j] = sn[j] - sp[j];
    float s_s[6];
#pragma unroll
    for (int i = 0; i < 6; ++i) {
      float acc = sf[i];
#pragma unroll
      for (int j = 0; j < 6; ++j) acc = fmaf(G[i][j], ds_[j], acc);
      s_s[i] = acc;
    }
    // Dm = P_next - Pp (reuse A);  T1 = G*Dm (into Pp);  P_s = P_f + T1*G^T
#pragma unroll
    for (int i = 0; i < 6; ++i)
#pragma unroll
      for (int j = 0; j < 6; ++j) A[i][j] = Pn[i][j] - Pp[i][j];
#pragma unroll
    for (int i = 0; i < 6; ++i)
#pragma unroll
      for (int j = 0; j < 6; ++j) {
        float acc = 0.0f;
#pragma unroll
        for (int k = 0; k < 6; ++k) acc = fmaf(G[i][k], A[k][j], acc);
        Pp[i][j] = acc;
      }
#pragma unroll
    for (int i = 0; i < 6; ++i)
#pragma unroll
      for (int j = 0; j < 6; ++j) {
        float acc = Pf[i][j];
#pragma unroll
        for (int k = 0; k < 6; ++k) acc = fmaf(Pp[i][k], G[j][k], acc);
        Pn[i][j] = acc;
      }
#pragma unroll
    for (int j = 0; j < 6; ++j) sn[j] = s_s[j];

    if (n >= 2) {
      const float z0 = zt[((size_t)n * 3 + 0) * B_N + b];
      const float z1 = zt[((size_t)n * 3 + 1) * B_N + b];
      const float z2 = zt[((size_t)n * 3 + 2) * B_N + b];
      loss += loss_term(z0, z1, z2, sn, Pn, R0, R1, R2);
    }
  }

  // ---- per-wave loss reduction through the matrix pipe ----
  float wave_sum;
#if HAVE_WMMA_F32X4
  {
    // A(16x4): lane L<16 -> (M=L,K=0)=loss ; lane L>=16 -> (M=L-16,K=2)=loss.
    // B(4x16) = ones  =>  D[M][N] = loss_M + loss_{M+16} for every N.
    v2f av;
    av.x = loss;
    av.y = 0.0f;
    v2f bv;
    bv.x = 1.0f;
    bv.y = 1.0f;
    v8f acc = {0.f, 0.f, 0.f, 0.f, 0.f, 0.f, 0.f, 0.f};
    acc = __builtin_amdgcn_wmma_f32_16x16x4_f32(false, av, false, bv, (short)0,
                                                acc, false, false);
    float q = acc[0] + acc[1] + acc[2] + acc[3] + acc[4] + acc[5] + acc[6] +
              acc[7];  // half-sum per 16-lane group
    wave_sum = q + __shfl_xor(q, 16, 32);
  }
#else
  {
    float q = loss;
#pragma unroll
    for (int off = 16; off > 0; off >>= 1) q += __shfl_xor(q, off, 32);
    wave_sum = q;
  }
#endif
  const int lane = threadIdx.x & 31;
  const int wave = (blockIdx.x * blockDim.x + threadIdx.x) >> 5;
  if (lane == 0) partials[wave] = wave_sum;
}

// Deterministic final reduction: 128 wave partials -> d_out[0].
__global__ __launch_bounds__(128) void reduce_partials_kernel(
    const float* __restrict__ partials, float* __restrict__ out) {
  __shared__ float sh[128];
  const int tid = threadIdx.x;
  sh[tid] = partials[tid];
  __syncthreads();
#pragma unroll
  for (int stp = 64; stp > 0; stp >>= 1) {
    if (tid < stp) sh[tid] += sh[tid + stp];
    __syncthreads();
  }
  if (tid == 0) out[0] = sh[0];
}

extern "C" void kernel_launch(void* const* d_in, const int* in_sizes, int n_in,
                              void* d_out, int out_size, void* d_ws,
                              size_t ws_size, hipStream_t stream) {
  (void)in_sizes;
  (void)n_in;
  (void)out_size;
  (void)ws_size;  // requires ~63 MB workspace
  const float* params = (const float*)d_in[0];        // (4,)
  const float* covp = (const float*)d_in[1];          // (7,)
  const float* init = (const float*)d_in[2];          // (B,6)
  const float* z = (const float*)d_in[3];             // (B,T,3)

  float* zt = (float*)d_ws;                                   // 384*B floats
  float* traj = zt + (size_t)T_N * 3 * B_N;                   // 27*T*B floats
  float* partials = traj + (size_t)27 * T_N * B_N;            // 128 floats

  transpose_z_kernel<<<dim3((T_N * 3) / 32, B_N / 32), dim3(32, 8), 0, stream>>>(
      z, zt);
  ekf_rts_kernel<<<B_N / 256, 256, 0, stream>>>(params, covp, init, zt, traj,
                                                partials);
  reduce_partials_kernel<<<1, 128, 0, stream>>>(partials, (float*)d_out);
}